// TrimmedMAELoss_31963146617385
// MI455X (gfx1250) — compile-verified
//
#include <hip/hip_runtime.h>
#include <stdint.h>

// Trimmed MAE via 2-level (16+16 bit) radix select on |pred-target| bit patterns.
// Memory-bound: two streaming passes over 118MB of inputs (~10us at 23.3 TB/s).
// wave32 / CDNA5: WMMA f32 16x16x4 used as the cross-lane reduction engine.

typedef float v2f __attribute__((ext_vector_type(2)));
typedef float v8f __attribute__((ext_vector_type(8)));

#define NBINS 65536
#define THREADS 256
#define PASS_BLOCKS 2400   // 2400*256 = 614400 threads; 9830400/4/614400 = 4 float4s per thread

// ---------------------------------------------------------------- zero workspace
__global__ void tmae_zero(unsigned* __restrict__ ws, int nwords) {
  int i = blockIdx.x * blockDim.x + threadIdx.x;
  if (i < nwords) ws[i] = 0u;
}

// ---------------------------------------------------------------- pass 1: coarse hist + M
__global__ void tmae_pass1(const float* __restrict__ p, const float* __restrict__ t,
                           const int* __restrict__ m, int N,
                           unsigned* __restrict__ hist1, unsigned* __restrict__ mcnt) {
  const int tid    = blockIdx.x * blockDim.x + threadIdx.x;
  const int stride = gridDim.x * blockDim.x;
  const int N4     = N >> 2;
  const float4* p4 = (const float4*)p;
  const float4* t4 = (const float4*)t;
  const int4*   m4 = (const int4*)m;

  unsigned localM = 0;
  for (int i = tid; i < N4; i += stride) {
    __builtin_prefetch(&p4[i + stride], 0, 1);
    __builtin_prefetch(&t4[i + stride], 0, 1);
    float4 pv = p4[i]; float4 tv = t4[i]; int4 mv = m4[i];
    if (mv.x) { unsigned u = __float_as_uint(fabsf(pv.x - tv.x)); atomicAdd(&hist1[u >> 16], 1u); localM++; }
    if (mv.y) { unsigned u = __float_as_uint(fabsf(pv.y - tv.y)); atomicAdd(&hist1[u >> 16], 1u); localM++; }
    if (mv.z) { unsigned u = __float_as_uint(fabsf(pv.z - tv.z)); atomicAdd(&hist1[u >> 16], 1u); localM++; }
    if (mv.w) { unsigned u = __float_as_uint(fabsf(pv.w - tv.w)); atomicAdd(&hist1[u >> 16], 1u); localM++; }
  }
  for (int i = (N4 << 2) + tid; i < N; i += stride) {
    if (m[i]) { unsigned u = __float_as_uint(fabsf(p[i] - t[i])); atomicAdd(&hist1[u >> 16], 1u); localM++; }
  }

  __shared__ unsigned sm[THREADS];
  sm[threadIdx.x] = localM;
  __syncthreads();
  for (int o = THREADS / 2; o > 0; o >>= 1) {
    if (threadIdx.x < o) sm[threadIdx.x] += sm[threadIdx.x + o];
    __syncthreads();
  }
  if (threadIdx.x == 0) atomicAdd(mcnt, sm[0]);
}

// ---------------------------------------------------------------- scan1: find coarse bin of k-th value
// ctrl: [0]=M, [1]=k, [2]=coarse bin b, [3]=count strictly below bin b
__global__ void tmae_scan1(const unsigned* __restrict__ hist1, unsigned* __restrict__ ctrl) {
  __shared__ unsigned csum[THREADS];
  const int t = threadIdx.x;
  unsigned s = 0;
  for (int j = 0; j < NBINS / THREADS; j++) s += hist1[t * (NBINS / THREADS) + j];
  csum[t] = s;
  __syncthreads();
  if (t == 0) {
    unsigned M = ctrl[0];
    float kf = (float)M * 0.8f;          // matches reference: f32(M) * f32(0.8), floor
    unsigned k = (unsigned)kf;
    unsigned b = 0, cum = 0;
    if (k > 0) {
      int c = 0;
      while (c < THREADS && cum + csum[c] < k) { cum += csum[c]; c++; }
      b = (unsigned)c * (NBINS / THREADS);
      while (b < NBINS) { unsigned h = hist1[b]; if (cum + h >= k) break; cum += h; b++; }
    }
    ctrl[1] = k; ctrl[2] = b; ctrl[3] = cum;
  }
}

// ---------------------------------------------------------------- pass 2: sum below b, fine hist at b
__global__ void tmae_pass2(const float* __restrict__ p, const float* __restrict__ t,
                           const int* __restrict__ m, int N,
                           const unsigned* __restrict__ ctrl,
                           unsigned* __restrict__ hist2, float* __restrict__ partials) {
  const unsigned b = ctrl[2];
  const int tid    = blockIdx.x * blockDim.x + threadIdx.x;
  const int stride = gridDim.x * blockDim.x;
  const int N4     = N >> 2;
  const float4* p4 = (const float4*)p;
  const float4* t4 = (const float4*)t;
  const int4*   m4 = (const int4*)m;

  float s = 0.0f;
  for (int i = tid; i < N4; i += stride) {
    __builtin_prefetch(&p4[i + stride], 0, 1);
    __builtin_prefetch(&t4[i + stride], 0, 1);
    float4 pv = p4[i]; float4 tv = t4[i]; int4 mv = m4[i];
    #define TMAE_DO(c)                                                                 \
      if (mv.c) {                                                                      \
        float r = fabsf(pv.c - tv.c); unsigned u = __float_as_uint(r);                 \
        unsigned hi = u >> 16;                                                         \
        if (hi < b) s += r;                                                            \
        else if (hi == b) atomicAdd(&hist2[u & 0xFFFFu], 1u);                          \
      }
    TMAE_DO(x) TMAE_DO(y) TMAE_DO(z) TMAE_DO(w)
    #undef TMAE_DO
  }
  for (int i = (N4 << 2) + tid; i < N; i += stride) {
    if (m[i]) {
      float r = fabsf(p[i] - t[i]); unsigned u = __float_as_uint(r);
      unsigned hi = u >> 16;
      if (hi < b) s += r;
      else if (hi == b) atomicAdd(&hist2[u & 0xFFFFu], 1u);
    }
  }

  // ---- wave32 reduction of s. All 256 threads converge here (no early returns),
  // so EXEC is all-ones as WMMA requires.
  float waveSum;
#if __has_builtin(__builtin_amdgcn_wmma_f32_16x16x4_f32)
  // A(16x4): VGPR0 = s per lane, VGPR1 = 0  =>  A[m][0]=s[m], A[m][2]=s[m+16].
  // B(4x16) = ones  =>  D[m][n] = s[m] + s[m+16] (rows replicated across n).
  // Lane-local sum of the 8 D VGPRs: lanes 0-15 get sum(s[0..7],s[16..23]),
  // lanes 16-31 get sum(s[8..15],s[24..31]); one xor-16 shuffle finishes it.
  v2f a;  a[0] = s;    a[1] = 0.0f;
  v2f bo; bo[0] = 1.0f; bo[1] = 1.0f;
  v8f d = {};
  d = __builtin_amdgcn_wmma_f32_16x16x4_f32(false, a, false, bo, (short)0, d, false, false);
  float w = d[0] + d[1] + d[2] + d[3] + d[4] + d[5] + d[6] + d[7];
  waveSum = w + __shfl_xor(w, 16, 32);
#else
  waveSum = s;
  for (int o = 16; o > 0; o >>= 1) waveSum += __shfl_xor(waveSum, o, 32);
#endif

  __shared__ float wsum[THREADS / 32];
  const int lane = threadIdx.x & 31;
  const int wave = threadIdx.x >> 5;
  if (lane == 0) wsum[wave] = waveSum;
  __syncthreads();
  if (threadIdx.x == 0) {
    float bs = 0.0f;
    for (int i = 0; i < THREADS / 32; i++) bs += wsum[i];
    partials[blockIdx.x] = bs;  // written unconditionally every call
  }
}

// ---------------------------------------------------------------- scan2: exact threshold + finalize
__global__ void tmae_scan2(const unsigned* __restrict__ hist2, const unsigned* __restrict__ ctrl,
                           const float* __restrict__ partials, int nPart, float* __restrict__ out) {
  __shared__ double pd[THREADS];
  __shared__ unsigned ccnt[THREADS];
  __shared__ double cws[THREADS];
  const int t = threadIdx.x;

  double s = 0.0;
  for (int i = t; i < nPart; i += THREADS) s += (double)partials[i];
  pd[t] = s;

  const unsigned b = ctrl[2];
  unsigned cnt = 0; double w = 0.0;
  for (int j = 0; j < NBINS / THREADS; j++) {
    unsigned idx = (unsigned)t * (NBINS / THREADS) + (unsigned)j;
    unsigned c = hist2[idx];
    cnt += c;
    w += (double)c * (double)__uint_as_float((b << 16) | idx);
  }
  ccnt[t] = cnt; cws[t] = w;
  __syncthreads();

  if (t == 0) {
    unsigned M = ctrl[0], k = ctrl[1], cum = ctrl[3];
    double acc = 0.0;
    for (int i = 0; i < THREADS; i++) acc += pd[i];   // sum of residuals in coarse bins < b
    float res = 0.0f;
    if (M > 0 && k > 0) {
      int c = 0;
      while (c < THREADS && cum + ccnt[c] < k) { cum += ccnt[c]; acc += cws[c]; c++; }
      if (c < THREADS) {
        for (unsigned j = (unsigned)c * (NBINS / THREADS); j < NBINS; j++) {
          unsigned h = hist2[j];
          double v = (double)__uint_as_float((b << 16) | j);
          if (cum + h >= k) { acc += (double)(k - cum) * v; break; }  // exact tie handling
          cum += h; acc += (double)h * v;
        }
      }
      res = (float)(acc / (2.0 * (double)M));
    }
    *out = res;
  }
}

// ---------------------------------------------------------------- launch
extern "C" void kernel_launch(void* const* d_in, const int* in_sizes, int n_in,
                              void* d_out, int out_size, void* d_ws, size_t ws_size,
                              hipStream_t stream) {
  const float* pred = (const float*)d_in[0];
  const float* targ = (const float*)d_in[1];
  const int*   mask = (const int*)d_in[2];
  float* out = (float*)d_out;
  const int N = in_sizes[0];

  // ws layout (u32 words): [0,65536) hist1 | [65536,131072) hist2 | [131072,131088) ctrl
  // then PASS_BLOCKS floats of block partials.
  unsigned* ws      = (unsigned*)d_ws;
  unsigned* hist1   = ws;
  unsigned* hist2   = ws + NBINS;
  unsigned* ctrl    = ws + 2 * NBINS;
  float*    partial = (float*)(ws + 2 * NBINS + 16);

  const int zeroWords = 2 * NBINS + 16;  // partials are overwritten unconditionally
  tmae_zero<<<(zeroWords + THREADS - 1) / THREADS, THREADS, 0, stream>>>(ws, zeroWords);
  tmae_pass1<<<PASS_BLOCKS, THREADS, 0, stream>>>(pred, targ, mask, N, hist1, &ctrl[0]);
  tmae_scan1<<<1, THREADS, 0, stream>>>(hist1, ctrl);
  tmae_pass2<<<PASS_BLOCKS, THREADS, 0, stream>>>(pred, targ, mask, N, ctrl, hist2, partial);
  tmae_scan2<<<1, THREADS, 0, stream>>>(hist2, ctrl, partial, PASS_BLOCKS, out);
}